// MoEModel_20796231647464
// MI455X (gfx1250) — compile-verified
//
#include <hip/hip_runtime.h>

typedef float v2f __attribute__((ext_vector_type(2)));
typedef float v8f __attribute__((ext_vector_type(8)));

#define BATCH 4096
#define IN_DIM 512
#define HID1 1024
#define HID2 512
#define NCLS 50
#define NEXP 8

// ---- CDNA5 async direct-to-LDS copy (GLOBAL_LOAD_ASYNC_TO_LDS_B128, ASYNCcnt) ----
__device__ __forceinline__ void async_load_b128(unsigned lds_off, const void* gaddr) {
    asm volatile("global_load_async_to_lds_b128 %0, %1, off"
                 :: "v"(lds_off), "v"(gaddr) : "memory");
}
__device__ __forceinline__ void wait_async0() {
    asm volatile("s_wait_asynccnt 0x0" ::: "memory");
}
__device__ __forceinline__ unsigned lds_off_of(const void* p) {
    // addrspace(3) offsets live in the low 32 bits of the generic address
    return (unsigned)(uintptr_t)p;
}

// ---------------- init: zero output region + expert counters ----------------
__global__ void k_init(float* __restrict__ out, int* __restrict__ cnt) {
    int i = blockIdx.x * blockDim.x + threadIdx.x;
    if (i < BATCH * NCLS) out[i] = 0.0f;
    if (i < NEXP) cnt[i] = 0;
}

// ---------------- router: one wave32 per token ----------------
__global__ __launch_bounds__(256) void k_router(
    const float* __restrict__ x, const float* __restrict__ Wr,
    const float* __restrict__ br, float* __restrict__ probs,
    float* __restrict__ gates, int* __restrict__ cnt, int* __restrict__ list) {
    int gid  = blockIdx.x * blockDim.x + threadIdx.x;
    int tok  = gid >> 5;
    int lane = gid & 31;
    if (tok >= BATCH) return;

    const float* xr = x + (size_t)tok * IN_DIM + lane * 16;
    float acc[NEXP];
#pragma unroll
    for (int e = 0; e < NEXP; ++e) acc[e] = 0.0f;
#pragma unroll
    for (int i = 0; i < 16; ++i) {
        float xv = xr[i];
        const float4 w0 = *(const float4*)(Wr + (size_t)(lane * 16 + i) * NEXP);
        const float4 w1 = *(const float4*)(Wr + (size_t)(lane * 16 + i) * NEXP + 4);
        acc[0] += xv * w0.x; acc[1] += xv * w0.y; acc[2] += xv * w0.z; acc[3] += xv * w0.w;
        acc[4] += xv * w1.x; acc[5] += xv * w1.y; acc[6] += xv * w1.z; acc[7] += xv * w1.w;
    }
#pragma unroll
    for (int off = 16; off > 0; off >>= 1) {
#pragma unroll
        for (int e = 0; e < NEXP; ++e) acc[e] += __shfl_xor(acc[e], off, 32);
    }
    if (lane == 0) {
        float p[NEXP];
        float mx = -1e30f;
#pragma unroll
        for (int e = 0; e < NEXP; ++e) { p[e] = acc[e] + br[e]; mx = fmaxf(mx, p[e]); }
        float s = 0.0f;
#pragma unroll
        for (int e = 0; e < NEXP; ++e) { p[e] = __expf(p[e] - mx); s += p[e]; }
        float inv = 1.0f / s;
#pragma unroll
        for (int e = 0; e < NEXP; ++e) { p[e] *= inv; probs[tok * NEXP + e] = p[e]; }
        // top-2, ties resolved toward lower index (matches jax.lax.top_k)
        int i1 = 0;
#pragma unroll
        for (int e = 1; e < NEXP; ++e) if (p[e] > p[i1]) i1 = e;
        int i2 = (i1 == 0) ? 1 : 0;
#pragma unroll
        for (int e = 0; e < NEXP; ++e) if (e != i1 && p[e] > p[i2]) i2 = e;
        float g[NEXP];
#pragma unroll
        for (int e = 0; e < NEXP; ++e) g[e] = 0.0f;
        g[i1] = 0.5f * p[i1];
        g[i2] = 0.5f * p[i2];
#pragma unroll
        for (int e = 0; e < NEXP; ++e) gates[tok * NEXP + e] = g[e];
        int q1 = atomicAdd(&cnt[i1], 1); list[i1 * BATCH + q1] = tok;
        int q2 = atomicAdd(&cnt[i2], 1); list[i2 * BATCH + q2] = tok;
    }
}

// ---------------- 8-entry exclusive prefix sum ----------------
__global__ void k_offsets(const int* __restrict__ cnt, int* __restrict__ offs) {
    if (blockIdx.x == 0 && threadIdx.x == 0) {
        int s = 0;
        for (int e = 0; e < NEXP; ++e) { offs[e] = s; s += cnt[e]; }
    }
}

// ---------------- gathered expert GEMM + bias + ReLU (layers 1 & 2) ----------------
// BM=64, BN=64, BK=32; 256 threads = 8 waves; wave tile 16(M) x 32(N), 2 accumulators.
// A staged via async direct-to-LDS (row-clamped: garbage rows feed D rows we never store).
// B staged transposed (BsT[col][k], stride 34) so B fragments are single ds_load_b64.
template <bool GATHER>
__global__ __launch_bounds__(256) void k_expert_gemm(
    const float* __restrict__ A, const float* __restrict__ W,
    const float* __restrict__ bias, float* __restrict__ Out,
    const int* __restrict__ cnt, const int* __restrict__ offs,
    const int* __restrict__ list, int K, int N, int lda) {
    constexpr int BM = 64, BN = 64, BK = 32;
    const int e  = blockIdx.z;
    const int c  = cnt[e];
    const int m0 = blockIdx.x * BM;
    if (m0 >= c) return;
    const int n0 = blockIdx.y * BN;

    __shared__ float As[BM][36];    // 144B row stride: 16B-aligned for b128, conflict-free frags
    __shared__ float BsT[BN][34];   // transposed; 8B-aligned b64 frags, conflict-free

    const int tid  = threadIdx.x;
    const int wv   = tid >> 5;
    const int lane = tid & 31;
    const int hl   = lane >> 4;     // k-half per ISA A/B fragment layout
    const int l16  = lane & 15;
    const int wm   = (wv & 3) * 16;
    const int wn   = (wv >> 2) * 32;
    const int gbase = offs[e];

    v8f acc0 = {0.f, 0.f, 0.f, 0.f, 0.f, 0.f, 0.f, 0.f};
    v8f acc1 = {0.f, 0.f, 0.f, 0.f, 0.f, 0.f, 0.f, 0.f};

    // A staging map: thread loads 2 consecutive float4 (32B) of one tile row per stage
    const int arow = (tid * 2) >> 3;          // 0..63
    const int ac   = (tid * 2) & 7;           // float4 slot (ac, ac+1)
    int r = m0 + arow; if (r > c - 1) r = c - 1;   // clamp: always a valid source row
    const size_t asrc = GATHER ? (size_t)list[e * BATCH + r] * (size_t)lda
                               : (size_t)(gbase + r) * (size_t)lda;
    const float*   gA     = A + asrc + ac * 4;
    const unsigned lds_a  = lds_off_of(&As[arow][ac * 4]);

    // B staging map: thread loads 2 consecutive float4 of one W row per stage
    const int bkr = (tid * 2) >> 4;           // k-row within tile (0..31)
    const int bc  = (tid * 2) & 15;           // float4 slot (bc, bc+1)
    const float* wp = W + (size_t)e * K * N + (size_t)bkr * N + n0 + bc * 4;

    for (int k0 = 0; k0 < K; k0 += BK) {
        // A: two async b128 transfers straight into LDS (ASYNCcnt)
        async_load_b128(lds_a,      gA + k0);
        async_load_b128(lds_a + 16, gA + k0 + 4);
        // B: global -> regs -> transposed LDS
        {
            const float4 w0 = *(const float4*)(wp + (size_t)k0 * N);
            const float4 w1 = *(const float4*)(wp + (size_t)k0 * N + 4);
            const int cb = bc * 4;
            BsT[cb + 0][bkr] = w0.x; BsT[cb + 1][bkr] = w0.y;
            BsT[cb + 2][bkr] = w0.z; BsT[cb + 3][bkr] = w0.w;
            BsT[cb + 4][bkr] = w1.x; BsT[cb + 5][bkr] = w1.y;
            BsT[cb + 6][bkr] = w1.z; BsT[cb + 7][bkr] = w1.w;
        }
        wait_async0();
        __syncthreads();
#pragma unroll
        for (int kk = 0; kk < BK; kk += 4) {
            const int ka = kk + hl * 2;       // lanes 0-15: k0,k1 ; lanes 16-31: k2,k3
            const v2f a  = *(const v2f*)(&As[wm + l16][ka]);
            const v2f b0 = *(const v2f*)(&BsT[wn + l16][ka]);
            const v2f b1 = *(const v2f*)(&BsT[wn + 16 + l16][ka]);
            acc0 = __builtin_amdgcn_wmma_f32_16x16x4_f32(false, a, false, b0, (short)0, acc0, false, false);
            acc1 = __builtin_amdgcn_wmma_f32_16x16x4_f32(false, a, false, b1, (short)0, acc1, false, false);
        }
        __syncthreads();
    }
    // epilogue: C/D layout — VGPR j holds row (j + hl*8), col = l16 within tile
    const int col0 = n0 + wn + l16;
    const int col1 = col0 + 16;
#pragma unroll
    for (int j = 0; j < 8; ++j) {
        int ri = m0 + wm + j + hl * 8;
        if (ri < c) {
            size_t orow = (size_t)(gbase + ri) * (size_t)N;
            float v0 = acc0[j] + bias[e * N + col0];
            float v1 = acc1[j] + bias[e * N + col1];
            Out[orow + col0] = fmaxf(v0, 0.0f);
            Out[orow + col1] = fmaxf(v1, 0.0f);
        }
    }
}

// ---------------- layer 3 (512 -> 50) + gated scatter-add into output ----------------
__global__ __launch_bounds__(256) void k_expert_out(
    const float* __restrict__ H, const float* __restrict__ W3,
    const float* __restrict__ b3, const float* __restrict__ gates,
    const int* __restrict__ cnt, const int* __restrict__ offs,
    const int* __restrict__ list, float* __restrict__ out) {
    constexpr int BM = 64, BK = 32, K = HID2;
    const int e  = blockIdx.z;
    const int c  = cnt[e];
    const int m0 = blockIdx.x * BM;
    if (m0 >= c) return;

    __shared__ float As[BM][36];
    __shared__ float BsT[64][34];

    const int tid  = threadIdx.x;
    const int wv   = tid >> 5;
    const int lane = tid & 31;
    const int hl   = lane >> 4;
    const int l16  = lane & 15;
    const int wm   = (wv & 3) * 16;
    const int wn   = (wv >> 2) * 32;
    const int gbase = offs[e];

    v8f acc0 = {0.f, 0.f, 0.f, 0.f, 0.f, 0.f, 0.f, 0.f};
    v8f acc1 = {0.f, 0.f, 0.f, 0.f, 0.f, 0.f, 0.f, 0.f};

    const int arow = (tid * 2) >> 3;
    const int ac   = (tid * 2) & 7;
    int r = m0 + arow; if (r > c - 1) r = c - 1;
    const float*   gA    = H + (size_t)(gbase + r) * HID2 + ac * 4;
    const unsigned lds_a = lds_off_of(&As[arow][ac * 4]);

    for (int k0 = 0; k0 < K; k0 += BK) {
        async_load_b128(lds_a,      gA + k0);
        async_load_b128(lds_a + 16, gA + k0 + 4);
        // B: 50 valid cols; guarded scalar loads/stores (cols >= 50 feed only
        // discarded D columns, so stale LDS there is harmless)
#pragma unroll
        for (int q = 0; q < 8; ++q) {
            int idx  = tid + q * 256;
            int brow = idx >> 6, bcol = idx & 63;
            if (bcol < NCLS)
                BsT[bcol][brow] = W3[(size_t)e * K * NCLS + (size_t)(k0 + brow) * NCLS + bcol];
        }
        wait_async0();
        __syncthreads();
#pragma unroll
        for (int kk = 0; kk < BK; kk += 4) {
            const int ka = kk + hl * 2;
            const v2f a  = *(const v2f*)(&As[wm + l16][ka]);
            const v2f b0 = *(const v2f*)(&BsT[wn + l16][ka]);
            const v2f b1 = *(const v2f*)(&BsT[wn + 16 + l16][ka]);
            acc0 = __builtin_amdgcn_wmma_f32_16x16x4_f32(false, a, false, b0, (short)0, acc0, false, false);
            acc1 = __builtin_amdgcn_wmma_f32_16x16x4_f32(false, a, false, b1, (short)0, acc1, false, false);
        }
        __syncthreads();
    }
    const int col0 = wn + l16;
    const int col1 = col0 + 16;
#pragma unroll
    for (int j = 0; j < 8; ++j) {
        int ri = m0 + wm + j + hl * 8;
        if (ri < c) {
            int   tok = list[e * BATCH + ri];
            float g   = gates[tok * NEXP + e];
            if (col0 < NCLS)
                atomicAdd(&out[(size_t)tok * NCLS + col0], g * (acc0[j] + b3[e * NCLS + col0]));
            if (col1 < NCLS)
                atomicAdd(&out[(size_t)tok * NCLS + col1], g * (acc1[j] + b3[e * NCLS + col1]));
        }
    }
}

// ---------------- host launcher ----------------
extern "C" void kernel_launch(void* const* d_in, const int* in_sizes, int n_in,
                              void* d_out, int out_size, void* d_ws, size_t ws_size,
                              hipStream_t stream) {
    (void)in_sizes; (void)n_in; (void)out_size; (void)ws_size;
    const float* x  = (const float*)d_in[0];
    const float* Wr = (const float*)d_in[1];
    const float* br = (const float*)d_in[2];
    const float* W1 = (const float*)d_in[3];
    const float* b1 = (const float*)d_in[4];
    const float* W2 = (const float*)d_in[5];
    const float* b2 = (const float*)d_in[6];
    const float* W3 = (const float*)d_in[7];
    const float* b3 = (const float*)d_in[8];

    float* out   = (float*)d_out;            // [4096 x 50]
    float* probs = out + BATCH * NCLS;       // [4096 x 8]

    char* ws = (char*)d_ws;
    constexpr size_t OFF_GATES = 0;                                          // f32[4096*8]
    constexpr size_t OFF_CNT   = OFF_GATES + (size_t)BATCH * NEXP * 4;       // int[8]
    constexpr size_t OFF_OFFS  = OFF_CNT + 64;                               // int[8]
    constexpr size_t OFF_LIST  = OFF_OFFS + 64;                              // int[8*4096]
    constexpr size_t OFF_H1    = (OFF_LIST + (size_t)NEXP * BATCH * 4 + 255) & ~(size_t)255;
    constexpr size_t OFF_H2    = OFF_H1 + (size_t)2 * BATCH * HID1 * 4;      // 8192 x 1024 f32
    float* gates = (float*)(ws + OFF_GATES);
    int*   cnt   = (int*)(ws + OFF_CNT);
    int*   offs  = (int*)(ws + OFF_OFFS);
    int*   list  = (int*)(ws + OFF_LIST);
    float* h1    = (float*)(ws + OFF_H1);                                    // compact [8192,1024]
    float* h2    = (float*)(ws + OFF_H2);                                    // compact [8192,512]

    k_init<<<(BATCH * NCLS + 255) / 256, 256, 0, stream>>>(out, cnt);
    k_router<<<(BATCH * 32) / 256, 256, 0, stream>>>(x, Wr, br, probs, gates, cnt, list);
    k_offsets<<<1, 32, 0, stream>>>(cnt, offs);

    dim3 g1(BATCH / 64, HID1 / 64, NEXP);
    k_expert_gemm<true><<<g1, 256, 0, stream>>>(x, W1, b1, h1, cnt, offs, list, IN_DIM, HID1, IN_DIM);

    dim3 g2(BATCH / 64, HID2 / 64, NEXP);
    k_expert_gemm<false><<<g2, 256, 0, stream>>>(h1, W2, b2, h2, cnt, offs, list, HID1, HID2, HID1);

    dim3 g3(BATCH / 64, 1, NEXP);
    k_expert_out<<<g3, 256, 0, stream>>>(h2, W3, b3, gates, cnt, offs, list, out);
}